// EGNN_61856118997065
// MI455X (gfx1250) — compile-verified
//
#include <hip/hip_runtime.h>
#include <math.h>

#define N_NODES 50000
#define N_EDGES 800000
#define D_FEAT  256
#define OUT_DIM 128
#define NBLK    ((N_NODES + 255) / 256)   // 196 scan blocks per count array

typedef __attribute__((ext_vector_type(2))) float v2f;
typedef __attribute__((ext_vector_type(8))) float v8f;

// ---------------------------------------------------------------------------
// Zero the integer count / fill arrays (both edge sets).
// ---------------------------------------------------------------------------
__global__ void k_init(int* __restrict__ cs1, int* __restrict__ cr1,
                       int* __restrict__ cs2, int* __restrict__ cr2,
                       int* __restrict__ f1,  int* __restrict__ f2) {
    int i = blockIdx.x * blockDim.x + threadIdx.x;
    if (i < N_NODES) {
        cs1[i] = 0; cr1[i] = 0; cs2[i] = 0; cr2[i] = 0; f1[i] = 0; f2[i] = 0;
    }
}

// ---------------------------------------------------------------------------
// Integer degree counts: one thread per edge (both edge sets in one grid).
// deg = cnt + 1 (self loop) is applied at use sites.
// ---------------------------------------------------------------------------
__global__ void k_count(const int* __restrict__ s1, const int* __restrict__ r1,
                        const int* __restrict__ s2, const int* __restrict__ r2,
                        int* __restrict__ cs1, int* __restrict__ cr1,
                        int* __restrict__ cs2, int* __restrict__ cr2) {
    int e = blockIdx.x * blockDim.x + threadIdx.x;
    if (e < N_EDGES) {
        atomicAdd(&cs1[s1[e]], 1);
        atomicAdd(&cr1[r1[e]], 1);
    } else if (e < 2 * N_EDGES) {
        int f = e - N_EDGES;
        atomicAdd(&cs2[s2[f]], 1);
        atomicAdd(&cr2[r2[f]], 1);
    }
}

// ---------------------------------------------------------------------------
// Exclusive scan of receiver counts -> CSR row offsets (two-level).
// ---------------------------------------------------------------------------
__device__ __forceinline__ int block_excl_scan_256(int v, int* sh) {
    const int t = threadIdx.x;
    sh[t] = v;
    __syncthreads();
    #pragma unroll
    for (int d = 1; d < 256; d <<= 1) {
        int x = (t >= d) ? sh[t - d] : 0;
        __syncthreads();
        sh[t] += x;
        __syncthreads();
    }
    return sh[t] - v;   // exclusive
}

// Level 1: per-256-chunk totals for cr1 (blocks 0..NBLK-1) and cr2 (NBLK..).
__global__ __launch_bounds__(256) void k_block_sums(
    const int* __restrict__ cr1, const int* __restrict__ cr2,
    int* __restrict__ bsum) {
    __shared__ int sh[256];
    const int bid = blockIdx.x;
    const int set = (bid >= NBLK);
    const int b   = set ? bid - NBLK : bid;
    const int* cr = set ? cr2 : cr1;
    const int i = b * 256 + threadIdx.x;
    sh[threadIdx.x] = (i < N_NODES) ? cr[i] : 0;
    __syncthreads();
    #pragma unroll
    for (int s = 128; s > 0; s >>= 1) {
        if (threadIdx.x < s) sh[threadIdx.x] += sh[threadIdx.x + s];
        __syncthreads();
    }
    if (threadIdx.x == 0) bsum[bid] = sh[0];
}

// Level 2: exclusive scan of the NBLK chunk totals (one block per edge set).
__global__ __launch_bounds__(256) void k_scan_top(
    const int* __restrict__ bsum, int* __restrict__ boff) {
    __shared__ int sh[256];
    const int set = blockIdx.x;
    const int t = threadIdx.x;
    int v = (t < NBLK) ? bsum[set * NBLK + t] : 0;
    int ex = block_excl_scan_256(v, sh);
    if (t < NBLK) boff[set * NBLK + t] = ex;
}

// Level 3: local exclusive scan + chunk offset -> final row offsets.
__global__ __launch_bounds__(256) void k_scan_final(
    const int* __restrict__ cr1, const int* __restrict__ cr2,
    const int* __restrict__ boff,
    int* __restrict__ off1, int* __restrict__ off2) {
    __shared__ int sh[256];
    const int bid = blockIdx.x;
    const int set = (bid >= NBLK);
    const int b   = set ? bid - NBLK : bid;
    const int* cr = set ? cr2 : cr1;
    int*      off = set ? off2 : off1;
    const int i = b * 256 + threadIdx.x;
    int v = (i < N_NODES) ? cr[i] : 0;
    int ex = block_excl_scan_256(v, sh);
    if (i < N_NODES) off[i] = ex + boff[bid];
}

// ---------------------------------------------------------------------------
// Bucket edges by receiver: eb[off[r] + slot] = sender.
// ---------------------------------------------------------------------------
__global__ void k_bucket(const int* __restrict__ s1, const int* __restrict__ r1,
                         const int* __restrict__ s2, const int* __restrict__ r2,
                         const int* __restrict__ off1, const int* __restrict__ off2,
                         int* __restrict__ f1, int* __restrict__ f2,
                         int* __restrict__ eb1, int* __restrict__ eb2) {
    int e = blockIdx.x * blockDim.x + threadIdx.x;
    if (e < N_EDGES) {
        int r = r1[e];
        int pos = off1[r] + atomicAdd(&f1[r], 1);
        eb1[pos] = s1[e];
    } else if (e < 2 * N_EDGES) {
        int f = e - N_EDGES;
        int r = r2[f];
        int pos = off2[r] + atomicAdd(&f2[r], 1);
        eb2[pos] = s2[f];
    }
}

// ---------------------------------------------------------------------------
// Fused dense update for both branches:
//   hs[:, 0:128]   = (nodes @ W1 + b1) * rsqrt(sdeg1)
//   hs[:, 128:256] = (nodes @ W2 + b2) * rsqrt(sdeg2)
// 512 threads = 16 waves; 16 rows of A in LDS (row pad 4 -> the 32-lane WMMA
// A-fetch covers all 64 LDS banks); waves 0-7 branch 1, waves 8-15 branch 2.
// K=256 via V_WMMA_F32_16X16X4_F32.
// ---------------------------------------------------------------------------
__global__ __launch_bounds__(512) void k_gemm12(
    const float* __restrict__ nodes,
    const float* __restrict__ W1, const float* __restrict__ b1,
    const float* __restrict__ W2, const float* __restrict__ b2,
    const int* __restrict__ cs1, const int* __restrict__ cs2,
    float* __restrict__ hs) {
    __shared__ float As[16][D_FEAT + 4];

    const int m0  = blockIdx.x * 16;
    const int tid = threadIdx.x;

    for (int i = tid; i < 16 * (D_FEAT / 4); i += 512) {
        int r  = i >> 6;
        int c4 = i & 63;
        float4 v = ((const float4*)(nodes + (size_t)(m0 + r) * D_FEAT))[c4];
        As[r][c4 * 4 + 0] = v.x; As[r][c4 * 4 + 1] = v.y;
        As[r][c4 * 4 + 2] = v.z; As[r][c4 * 4 + 3] = v.w;
    }
    __syncthreads();

    const int wave   = tid >> 5;
    const int lane   = tid & 31;
    const int half   = lane >> 4;
    const int l16    = lane & 15;
    const int branch = wave >> 3;
    const int n0     = (wave & 7) * 16;

    const float* __restrict__ W    = branch ? W2  : W1;
    const float* __restrict__ bias = branch ? b2  : b1;
    const int*   __restrict__ cs   = branch ? cs2 : cs1;

    v8f acc = {};
    #pragma unroll 8
    for (int kb = 0; kb < D_FEAT; kb += 4) {
        const int k0 = kb + 2 * half;
        v2f a, b;
        a.x = As[l16][k0];
        a.y = As[l16][k0 + 1];
        b.x = W[(size_t)k0 * OUT_DIM + n0 + l16];
        b.y = W[(size_t)(k0 + 1) * OUT_DIM + n0 + l16];
        acc = __builtin_amdgcn_wmma_f32_16x16x4_f32(false, a, false, b,
                                                    (short)0, acc, false, false);
    }

    const float bv = bias[n0 + l16];
    #pragma unroll
    for (int i = 0; i < 8; ++i) {
        const int row = m0 + i + 8 * half;
        const float v = (acc[i] + bv) * rsqrtf((float)(cs[row] + 1));
        hs[(size_t)row * (2 * OUT_DIM) + (size_t)branch * OUT_DIM + n0 + l16] = v;
    }
}

// ---------------------------------------------------------------------------
// Pull-mode aggregation: one wave32 per (node, branch). acc starts at the
// self-edge row hs[n]; loop over CSR in-neighbors gathering 512 B rows as
// float4/lane (L2-resident working set on the 192 MB L2); epilogue applies
// rsqrt(rdeg) and stores the final x matrix. Zero floating-point atomics.
// ---------------------------------------------------------------------------
__global__ __launch_bounds__(256) void k_gather(
    const float* __restrict__ hs,
    const int* __restrict__ off1, const int* __restrict__ off2,
    const int* __restrict__ cr1,  const int* __restrict__ cr2,
    const int* __restrict__ eb1,  const int* __restrict__ eb2,
    float* __restrict__ xs) {
    const int lane = threadIdx.x & 31;
    const int task = blockIdx.x * 8 + (threadIdx.x >> 5);
    if (task >= 2 * N_NODES) return;

    const int set = (task >= N_NODES);
    const int n   = set ? task - N_NODES : task;
    const int* __restrict__ off = set ? off2 : off1;
    const int* __restrict__ cnt = set ? cr2  : cr1;
    const int* __restrict__ eb  = set ? eb2  : eb1;

    const size_t col = (size_t)set * OUT_DIM;
    float4 acc = ((const float4*)(hs + (size_t)n * (2 * OUT_DIM) + col))[lane];

    const int st = off[n];
    const int c  = cnt[n];
    const int en = st + c;
    for (int j = st; j < en; ++j) {
        const int s = eb[j];
        const float4 v = ((const float4*)(hs + (size_t)s * (2 * OUT_DIM) + col))[lane];
        acc.x += v.x; acc.y += v.y; acc.z += v.z; acc.w += v.w;
    }

    const float sc = rsqrtf((float)(c + 1));
    float4 o;
    o.x = acc.x * sc; o.y = acc.y * sc; o.z = acc.z * sc; o.w = acc.w * sc;
    ((float4*)(xs + (size_t)n * (2 * OUT_DIM) + col))[lane] = o;
}

// ---------------------------------------------------------------------------
// Final: out = relu(xs @ W3 + b3). 256 threads = 8 waves, one 16-col N-tile
// per wave; xs is already fully normalized.
// ---------------------------------------------------------------------------
__global__ __launch_bounds__(256) void k_gemm3(
    const float* __restrict__ xs,
    const float* __restrict__ W3, const float* __restrict__ b3,
    float* __restrict__ out) {
    __shared__ float As[16][2 * OUT_DIM + 4];

    const int m0  = blockIdx.x * 16;
    const int tid = threadIdx.x;

    for (int i = tid; i < 16 * 64; i += 256) {
        int r  = i >> 6;
        int c4 = i & 63;
        float4 v = ((const float4*)(xs + (size_t)(m0 + r) * (2 * OUT_DIM)))[c4];
        As[r][c4 * 4 + 0] = v.x; As[r][c4 * 4 + 1] = v.y;
        As[r][c4 * 4 + 2] = v.z; As[r][c4 * 4 + 3] = v.w;
    }
    __syncthreads();

    const int wave = tid >> 5;
    const int lane = tid & 31;
    const int half = lane >> 4;
    const int l16  = lane & 15;
    const int n0   = wave * 16;

    v8f acc = {};
    #pragma unroll 8
    for (int kb = 0; kb < 2 * OUT_DIM; kb += 4) {
        const int k0 = kb + 2 * half;
        v2f a, b;
        a.x = As[l16][k0];
        a.y = As[l16][k0 + 1];
        b.x = W3[(size_t)k0 * OUT_DIM + n0 + l16];
        b.y = W3[(size_t)(k0 + 1) * OUT_DIM + n0 + l16];
        acc = __builtin_amdgcn_wmma_f32_16x16x4_f32(false, a, false, b,
                                                    (short)0, acc, false, false);
    }

    const float bv = b3[n0 + l16];
    #pragma unroll
    for (int i = 0; i < 8; ++i) {
        const int row = m0 + i + 8 * half;
        out[(size_t)row * OUT_DIM + n0 + l16] = fmaxf(acc[i] + bv, 0.0f);
    }
}

// ---------------------------------------------------------------------------
extern "C" void kernel_launch(void* const* d_in, const int* in_sizes, int n_in,
                              void* d_out, int out_size, void* d_ws, size_t ws_size,
                              hipStream_t stream) {
    (void)in_sizes; (void)n_in; (void)out_size; (void)ws_size;

    const float* nodes = (const float*)d_in[0];
    const int*   s1    = (const int*)d_in[1];
    const int*   r1    = (const int*)d_in[2];
    const int*   s2    = (const int*)d_in[3];
    const int*   r2    = (const int*)d_in[4];
    const float* W1    = (const float*)d_in[5];
    const float* b1    = (const float*)d_in[6];
    const float* W2    = (const float*)d_in[7];
    const float* b2    = (const float*)d_in[8];
    const float* W3    = (const float*)d_in[9];
    const float* b3    = (const float*)d_in[10];
    float*       out   = (float*)d_out;

    // Workspace layout: big float4-aligned arrays first, then int arrays.
    char* p = (char*)d_ws;
    float* hs  = (float*)p; p += sizeof(float) * (size_t)N_NODES * (2 * OUT_DIM);
    float* xs  = (float*)p; p += sizeof(float) * (size_t)N_NODES * (2 * OUT_DIM);
    int*   eb1 = (int*)p;   p += sizeof(int) * (size_t)N_EDGES;
    int*   eb2 = (int*)p;   p += sizeof(int) * (size_t)N_EDGES;
    int*   cs1 = (int*)p;   p += sizeof(int) * N_NODES;
    int*   cr1 = (int*)p;   p += sizeof(int) * N_NODES;
    int*   cs2 = (int*)p;   p += sizeof(int) * N_NODES;
    int*   cr2 = (int*)p;   p += sizeof(int) * N_NODES;
    int*   f1  = (int*)p;   p += sizeof(int) * N_NODES;
    int*   f2  = (int*)p;   p += sizeof(int) * N_NODES;
    int*   off1 = (int*)p;  p += sizeof(int) * N_NODES;
    int*   off2 = (int*)p;  p += sizeof(int) * N_NODES;
    int*   bsum = (int*)p;  p += sizeof(int) * (2 * NBLK);
    int*   boff = (int*)p;

    k_init<<<(N_NODES + 255) / 256, 256, 0, stream>>>(cs1, cr1, cs2, cr2, f1, f2);

    k_count<<<(2 * N_EDGES + 255) / 256, 256, 0, stream>>>(
        s1, r1, s2, r2, cs1, cr1, cs2, cr2);

    k_block_sums<<<2 * NBLK, 256, 0, stream>>>(cr1, cr2, bsum);
    k_scan_top<<<2, 256, 0, stream>>>(bsum, boff);
    k_scan_final<<<2 * NBLK, 256, 0, stream>>>(cr1, cr2, boff, off1, off2);

    k_bucket<<<(2 * N_EDGES + 255) / 256, 256, 0, stream>>>(
        s1, r1, s2, r2, off1, off2, f1, f2, eb1, eb2);

    k_gemm12<<<N_NODES / 16, 512, 0, stream>>>(
        nodes, W1, b1, W2, b2, cs1, cs2, hs);

    k_gather<<<(2 * N_NODES) / 8, 256, 0, stream>>>(
        hs, off1, off2, cr1, cr2, eb1, eb2, xs);

    k_gemm3<<<N_NODES / 16, 256, 0, stream>>>(xs, W3, b3, out);
}